// SAGE_39350490366323
// MI455X (gfx1250) — compile-verified
//
#include <hip/hip_runtime.h>
#include <math.h>

typedef __attribute__((ext_vector_type(2))) float v2f;
typedef __attribute__((ext_vector_type(8))) float v8f;

#define HID  256
#define OUTC 41

// ---- CDNA5 async global->LDS copy helpers (ASYNCcnt-tracked) --------------
__device__ __forceinline__ void async_copy_b32(unsigned lds_byte_off,
                                               const void* gptr) {
    asm volatile("global_load_async_to_lds_b32 %0, %1, off"
                 :: "v"(lds_byte_off),
                    "v"((unsigned long long)(uintptr_t)gptr)
                 : "memory");
}
__device__ __forceinline__ void async_copy_b128(unsigned lds_byte_off,
                                                const void* gptr) {
    asm volatile("global_load_async_to_lds_b128 %0, %1, off"
                 :: "v"(lds_byte_off),
                    "v"((unsigned long long)(uintptr_t)gptr)
                 : "memory");
}
__device__ __forceinline__ void wait_async0() {
#if __has_builtin(__builtin_amdgcn_s_wait_asynccnt)
    __builtin_amdgcn_s_wait_asynccnt(0);
#else
    asm volatile("s_wait_asynccnt 0x0" ::: "memory");
#endif
}

__global__ void zero_f32(float* __restrict__ p, int n) {
    int i = blockIdx.x * blockDim.x + threadIdx.x;
    if (i < n) p[i] = 0.0f;
}

// C[M,N] = A[M,K] @ B[K,N], all fp32, via V_WMMA_F32_16X16X4_F32 (wave32).
// Block: 256 threads = 8 waves; block tile 64(M) x 64(N); each wave owns a
// 16-row strip and two 16x16 N-subtiles. K staged through LDS in steps of 16.
// Interior tiles stage LDS with GLOBAL_LOAD_ASYNC_TO_LDS (B32 transposed
// scatter for A, B128 for B); tails fall back to checked scalar loads.
__global__ __launch_bounds__(256)
void gemm_f32_wmma(const float* __restrict__ A, const float* __restrict__ B,
                   float* __restrict__ C, int M, int N, int K,
                   int lda, int ldb, int ldc) {
    __shared__ float As[16][65];   // [k][m] transposed; pad 65 -> conflict-free frags
    __shared__ float Bs[16][68];   // [k][n]; pad 68 -> conflict-free frags, 16B-aligned rows

    const int tid  = threadIdx.x;
    const int wave = tid >> 5;
    const int lane = tid & 31;
    const int hs   = lane >> 4;    // half-select: K pair {0,1} vs {2,3}
    const int l15  = lane & 15;

    const int bm = blockIdx.y * 64;
    const int bn = blockIdx.x * 64;
    const int wm = (wave & 3) * 16;
    const int wn = (wave >> 2) * 32;

    // LDS-relative byte addresses (generic ptr low 32 bits == LDS offset)
    const unsigned as_base = (unsigned)(uintptr_t)&As[0][0];
    const unsigned bs_base = (unsigned)(uintptr_t)&Bs[0][0];

    const bool fullMN = (bm + 64 <= M) && (bn + 64 <= N);

    // per-thread staging coordinates
    const int am = tid >> 4;        // 0..15 (A row group base; +16 per i)
    const int ak = tid & 15;        // 0..15 (A k within step)
    const int bk = tid >> 4;        // 0..15 (B k within step)
    const int bq = (tid & 15) * 4;  // 0..60 (B n quad)

    v8f acc0 = {};
    v8f acc1 = {};

    for (int k0 = 0; k0 < K; k0 += 16) {
        if (fullMN && (k0 + 16 <= K)) {
            // ---- fast path: async copies, no bounds checks ----
            #pragma unroll
            for (int i = 0; i < 4; ++i) {
                int m = am + i * 16;                 // 0..63
                async_copy_b32(as_base + (unsigned)(ak * 65 + m) * 4u,
                               A + (size_t)(bm + m) * lda + (k0 + ak));
            }
            async_copy_b128(bs_base + (unsigned)(bk * 68 + bq) * 4u,
                            B + (size_t)(k0 + bk) * ldb + (bn + bq));
            wait_async0();
        } else {
            // ---- checked path (K tail / small-N GEMMs) ----
            #pragma unroll
            for (int i = 0; i < 4; ++i) {
                int idx = tid + i * 256;
                int m = idx >> 4, k = idx & 15;
                int gm = bm + m, gk = k0 + k;
                As[k][m] = (gm < M && gk < K) ? A[(size_t)gm * lda + gk] : 0.0f;
            }
            #pragma unroll
            for (int i = 0; i < 4; ++i) {
                int idx = tid + i * 256;
                int k = idx >> 6, n = idx & 63;
                int gk = k0 + k, gn = bn + n;
                Bs[k][n] = (gk < K && gn < N) ? B[(size_t)gk * ldb + gn] : 0.0f;
            }
        }
        __syncthreads();

        #pragma unroll
        for (int kk = 0; kk < 16; kk += 4) {
            // A 16x4 fp32 fragment: lane L holds M=L&15, K = 2*hs + {0,1}
            v2f a;
            a.x = As[kk + 2 * hs + 0][wm + l15];
            a.y = As[kk + 2 * hs + 1][wm + l15];
            // B 4x16 fp32 fragment: lane L holds N=L&15, K = 2*hs + {0,1}
            v2f b0, b1;
            b0.x = Bs[kk + 2 * hs + 0][wn + l15];
            b0.y = Bs[kk + 2 * hs + 1][wn + l15];
            b1.x = Bs[kk + 2 * hs + 0][wn + 16 + l15];
            b1.y = Bs[kk + 2 * hs + 1][wn + 16 + l15];
            acc0 = __builtin_amdgcn_wmma_f32_16x16x4_f32(
                false, a, false, b0, (short)0, acc0, false, false);
            acc1 = __builtin_amdgcn_wmma_f32_16x16x4_f32(
                false, a, false, b1, (short)0, acc1, false, false);
        }
        __syncthreads();
    }

    // C/D 16x16 f32 layout: lane L, vgpr v -> row v + 8*(L>>4), col L&15
    #pragma unroll
    for (int v = 0; v < 8; ++v) {
        int gm  = bm + wm + v + 8 * hs;
        int gn0 = bn + wn + l15;
        int gn1 = gn0 + 16;
        if (gm < M) {
            if (gn0 < N) C[(size_t)gm * ldc + gn0] = acc0[v];
            if (gn1 < N) C[(size_t)gm * ldc + gn1] = acc1[v];
        }
    }
}

// One wave per edge: gather P[src] (256 ch), atomic-add into agg[dst] (L2-resident).
__global__ void scatter_mean0(const float* __restrict__ P,
                              const int* __restrict__ src,
                              const int* __restrict__ dst,
                              float* __restrict__ agg, float* __restrict__ cnt,
                              int E) {
    int gid  = blockIdx.x * blockDim.x + threadIdx.x;
    int e    = gid >> 5;
    int lane = gid & 31;
    if (e >= E) return;
    int s = src[e], d = dst[e];
    const float* p = P + (size_t)s * HID;
    float*       a = agg + (size_t)d * HID;
    #pragma unroll
    for (int c = lane; c < HID; c += 32) atomicAdd(&a[c], p[c]);
    if (lane == 0) atomicAdd(&cnt[d], 1.0f);
}

// h = relu(agg/cnt + R + bias), written in place over agg.
__global__ void combine_relu(float* __restrict__ agg, const float* __restrict__ R,
                             const float* __restrict__ cnt,
                             const float* __restrict__ bias, int rows) {
    int i = blockIdx.x * blockDim.x + threadIdx.x;
    if (i >= rows * HID) return;
    int r = i >> 8;              // HID == 256
    int c = i & (HID - 1);
    float k = fmaxf(cnt[r], 1.0f);
    float h = agg[i] / k + R[i] + bias[c];
    agg[i] = fmaxf(h, 0.0f);
}

// One wave per edge, 41 channels.
__global__ void scatter_mean1(const float* __restrict__ Q,
                              const int* __restrict__ src,
                              const int* __restrict__ dst,
                              float* __restrict__ agg, float* __restrict__ cnt,
                              int E) {
    int gid  = blockIdx.x * blockDim.x + threadIdx.x;
    int e    = gid >> 5;
    int lane = gid & 31;
    if (e >= E) return;
    int s = src[e], d = dst[e];
    const float* q = Q + (size_t)s * OUTC;
    float*       a = agg + (size_t)d * OUTC;
    for (int c = lane; c < OUTC; c += 32) atomicAdd(&a[c], q[c]);
    if (lane == 0) atomicAdd(&cnt[d], 1.0f);
}

// One wave per row: o = agg/cnt + S + bias; out = o - max - log(sum(exp)).
__global__ void finalize_logsoftmax(const float* __restrict__ agg,
                                    const float* __restrict__ S,
                                    const float* __restrict__ cnt,
                                    const float* __restrict__ bias,
                                    float* __restrict__ out, int rows) {
    int gid  = blockIdx.x * blockDim.x + threadIdx.x;
    int r    = gid >> 5;
    int lane = gid & 31;
    if (r >= rows) return;
    float k = fmaxf(cnt[r], 1.0f);
    int c0 = lane, c1 = lane + 32;
    float o0 = -INFINITY, o1 = -INFINITY;
    if (c0 < OUTC) o0 = agg[(size_t)r * OUTC + c0] / k + S[(size_t)r * OUTC + c0] + bias[c0];
    if (c1 < OUTC) o1 = agg[(size_t)r * OUTC + c1] / k + S[(size_t)r * OUTC + c1] + bias[c1];
    float m = fmaxf(o0, o1);
    #pragma unroll
    for (int off = 16; off; off >>= 1) m = fmaxf(m, __shfl_xor(m, off, 32));
    float s = 0.0f;
    if (c0 < OUTC) s += __expf(o0 - m);
    if (c1 < OUTC) s += __expf(o1 - m);
    #pragma unroll
    for (int off = 16; off; off >>= 1) s += __shfl_xor(s, off, 32);
    float lse = __logf(s);
    if (c0 < OUTC) out[(size_t)r * OUTC + c0] = o0 - m - lse;
    if (c1 < OUTC) out[(size_t)r * OUTC + c1] = o1 - m - lse;
}

extern "C" void kernel_launch(void* const* d_in, const int* in_sizes, int n_in,
                              void* d_out, int out_size, void* d_ws, size_t ws_size,
                              hipStream_t stream) {
    (void)n_in; (void)out_size; (void)ws_size;
    const float* x   = (const float*)d_in[0];
    const int*   ei0 = (const int*)d_in[1];
    const int*   ei1 = (const int*)d_in[2];
    const float* Wl0 = (const float*)d_in[5];
    const float* bl0 = (const float*)d_in[6];
    const float* Wr0 = (const float*)d_in[7];
    const float* Wl1 = (const float*)d_in[8];
    const float* bl1 = (const float*)d_in[9];
    const float* Wr1 = (const float*)d_in[10];

    const int E0 = in_sizes[1] / 2;   // 600000
    const int E1 = in_sizes[2] / 2;   // 102400
    const int N1 = 40000, N2 = 4096, INC = 602;

    // workspace layout (fp32):
    float* buf0 = (float*)d_ws;                 // P, then reused for R  [N1 x 256]
    float* buf1 = buf0 + (size_t)N1 * HID;      // agg0, then h in place [N1 x 256]
    float* cnt0 = buf1 + (size_t)N1 * HID;      // [N1]
    float* Q    = cnt0 + N1;                    // [N2 x 41]
    float* S    = Q + (size_t)N2 * OUTC;        // [N2 x 41]
    float* agg1 = S + (size_t)N2 * OUTC;        // [N2 x 41]
    float* cnt1 = agg1 + (size_t)N2 * OUTC;     // [N2]

    // zero accumulators (fresh every call; ws is not re-poisoned by harness)
    { int n = N1 * HID;  zero_f32<<<(n + 255) / 256, 256, 0, stream>>>(buf1, n); }
    zero_f32<<<(N1 + 255) / 256, 256, 0, stream>>>(cnt0, N1);
    { int n = N2 * OUTC; zero_f32<<<(n + 255) / 256, 256, 0, stream>>>(agg1, n); }
    zero_f32<<<(N2 + 255) / 256, 256, 0, stream>>>(cnt1, N2);

    dim3 blk(256);
    // P = x[:N1] @ Wl0
    dim3 g0((HID + 63) / 64, (N1 + 63) / 64);
    gemm_f32_wmma<<<g0, blk, 0, stream>>>(x, Wl0, buf0, N1, HID, INC, INC, HID, HID);
    // agg0 += P[src] per edge, cnt0 per dst
    { int t = E0 * 32;
      scatter_mean0<<<(t + 255) / 256, 256, 0, stream>>>(buf0, ei0, ei0 + E0, buf1, cnt0, E0); }
    // R = x[:N1] @ Wr0 (overwrites P — stream-ordered after scatter)
    gemm_f32_wmma<<<g0, blk, 0, stream>>>(x, Wr0, buf0, N1, HID, INC, INC, HID, HID);
    // h = relu(agg0/cnt0 + R + bl0), in place into buf1
    { int n = N1 * HID;
      combine_relu<<<(n + 255) / 256, 256, 0, stream>>>(buf1, buf0, cnt0, bl0, N1); }

    // layer 1
    dim3 g1((OUTC + 63) / 64, (N2 + 63) / 64);
    gemm_f32_wmma<<<g1, blk, 0, stream>>>(buf1, Wl1, Q, N2, OUTC, HID, HID, OUTC, OUTC);
    { int t = E1 * 32;
      scatter_mean1<<<(t + 255) / 256, 256, 0, stream>>>(Q, ei1, ei1 + E1, agg1, cnt1, E1); }
    gemm_f32_wmma<<<g1, blk, 0, stream>>>(buf1, Wr1, S, N2, OUTC, HID, HID, OUTC, OUTC);
    { int t = N2 * 32;
      finalize_logsoftmax<<<(t + 255) / 256, 256, 0, stream>>>(agg1, S, cnt1, bl1,
                                                               (float*)d_out, N2); }
}